// CapsuleLayer_22239340659091
// MI455X (gfx1250) — compile-verified
//
#include <hip/hip_runtime.h>
#include <hip/hip_bf16.h>
#include <math.h>

// Problem constants (from reference)
#define BATCH   64
#define NI      2048
#define DIN     8
#define NK      16      // num_capsule
#define DOUT    16      // dim_capsule
#define EPSV    1e-7f

// Tiling
#define BT      16          // batch tile (WMMA M)
#define CHUNK_I 16          // i's per workgroup
#define TILE_I  2           // i's per WMMA phase
#define USTRIDE 260         // padded row stride (floats) for U in LDS (bank-conflict free D stores)
#define UILEN   (16*USTRIDE)

typedef __attribute__((ext_vector_type(2))) float v2f;
typedef __attribute__((ext_vector_type(8))) float v8f;

__global__ void caps_zero(float* __restrict__ s) {
    int i = blockIdx.x * blockDim.x + threadIdx.x;
    if (i < BATCH * NK * DOUT) s[i] = 0.0f;
}

// One routing pass: recompute u_hat via fp32 WMMA, compute coupling c, accumulate s.
// ROUTED=false: c = 1/16 (softmax of zero logits). ROUTED=true: logits = u_hat . vacc.
template <bool ROUTED>
__global__ __launch_bounds__(256) void caps_pass(const float* __restrict__ X,   // [B, I, DIN]
                                                 const float* __restrict__ Wp,  // [I, K, DIN, DOUT]
                                                 const float* __restrict__ vacc,// [B, K, DOUT] (sum of v's)
                                                 float* __restrict__ s_out) {   // [B, K, DOUT] accumulated
    __shared__ float x_lds[BT * CHUNK_I * DIN];        // 2048 floats
    __shared__ float u_lds[TILE_I * UILEN];            // 8320 floats

    const int tid  = threadIdx.x;
    const int b0   = blockIdx.y * BT;
    const int i0   = blockIdx.x * CHUNK_I;

    // ---- Stage X chunk into LDS (coalesced over (i,d) within each b row) ----
    for (int idx = tid; idx < BT * CHUNK_I * DIN; idx += 256) {
        int b_l = idx / (CHUNK_I * DIN);
        int rem = idx - b_l * (CHUNK_I * DIN);
        int i_l = rem >> 3;
        int d   = rem & 7;
        x_lds[(i_l * BT + b_l) * DIN + d] =
            X[((size_t)(b0 + b_l) * NI + (i0 + i_l)) * DIN + d];
    }

    // ---- Per-thread routing state ----
    const int b_l = tid >> 4;      // 0..15 (lanes: 2*wave + half)
    const int kk  = tid & 15;      // capsule k == lane%16 -> 16-lane shfl reductions
    float vreg[DOUT];
    if (ROUTED) {
#pragma unroll
        for (int e = 0; e < DOUT; ++e)
            vreg[e] = vacc[(((b0 + b_l) * NK) + kk) * DOUT + e];
    }
    float acc[DOUT];
#pragma unroll
    for (int e = 0; e < DOUT; ++e) acc[e] = 0.0f;

    // WMMA lane decomposition
    const int wv   = tid >> 5;     // wave 0..7 -> tiles (capsules) 2wv, 2wv+1
    const int ln   = tid & 31;
    const int half = ln >> 4;
    const int lm   = ln & 15;

    __syncthreads();

    for (int ib = 0; ib < CHUNK_I; ib += TILE_I) {
        // ================= WMMA phase: U[i, b, k*16+e] for TILE_I i's =================
#pragma unroll
        for (int ii = 0; ii < TILE_I; ++ii) {
            const int i_l = ib + ii;
            const int i_g = i0 + i_l;
            v8f c0 = {};
            v8f c1 = {};
#pragma unroll
            for (int ks = 0; ks < 2; ++ks) {            // two K=4 steps cover DIN=8
                const int d0 = 4 * ks + 2 * half;
                // A: X[b=lm, d0..d0+1]  (16x4 fp32 A layout)
                v2f a = *(const v2f*)&x_lds[(i_l * BT + lm) * DIN + d0];
                // B: W[i, k=tile, d0..d0+1, e=lm]  (4x16 fp32 B layout)
                size_t baseA = (((size_t)i_g * NK + 2 * wv) * DIN + d0) * DOUT + lm;
                size_t baseB = (((size_t)i_g * NK + 2 * wv + 1) * DIN + d0) * DOUT + lm;
                v2f bA, bB;
                bA.x = Wp[baseA];  bA.y = Wp[baseA + DOUT];
                bB.x = Wp[baseB];  bB.y = Wp[baseB + DOUT];
                c0 = __builtin_amdgcn_wmma_f32_16x16x4_f32(false, a, false, bA,
                                                           (short)0, c0, false, false);
                c1 = __builtin_amdgcn_wmma_f32_16x16x4_f32(false, a, false, bB,
                                                           (short)0, c1, false, false);
            }
            // D layout: vgpr r -> row M = r + 8*half; col N = lane%16
            const int colA = (2 * wv) * DOUT + lm;
#pragma unroll
            for (int r = 0; r < 8; ++r) {
                const int M = r + 8 * half;
                u_lds[ii * UILEN + M * USTRIDE + colA]        = c0[r];
                u_lds[ii * UILEN + M * USTRIDE + colA + DOUT] = c1[r];
            }
        }
        __syncthreads();

        // ================= Routing phase (VALU) =================
#pragma unroll
        for (int ii = 0; ii < TILE_I; ++ii) {
            const float* up = &u_lds[ii * UILEN + b_l * USTRIDE + kk * DOUT];
            float u[DOUT];
#pragma unroll
            for (int e = 0; e < DOUT; ++e) u[e] = up[e];

            float c;
            if (ROUTED) {
                float bij = 0.0f;
#pragma unroll
                for (int e = 0; e < DOUT; ++e) bij = fmaf(u[e], vreg[e], bij);
                // softmax over the 16 capsules (lanes 0..15 / 16..31 hold k=0..15)
                float m = bij;
#pragma unroll
                for (int msk = 1; msk < 16; msk <<= 1)
                    m = fmaxf(m, __shfl_xor(m, msk, 32));
                float p = __expf(bij - m);
                float sum = p;
#pragma unroll
                for (int msk = 1; msk < 16; msk <<= 1)
                    sum += __shfl_xor(sum, msk, 32);
                c = p / sum;
            } else {
                c = 0.0625f;   // softmax(0) over 16 capsules
            }
#pragma unroll
            for (int e = 0; e < DOUT; ++e) acc[e] = fmaf(c, u[e], acc[e]);
        }
        __syncthreads();
    }

    // ---- Accumulate partial s into global (tiny 64KB buffer, L2-resident atomics) ----
    float* sp = &s_out[(((b0 + b_l) * NK) + kk) * DOUT];
#pragma unroll
    for (int e = 0; e < DOUT; ++e) atomicAdd(&sp[e], acc[e]);
}

// v = squash(s); maintain vacc; zero s for next pass; final round writes d_out.
__global__ void caps_squash(float* __restrict__ s, float* __restrict__ vacc,
                            float* __restrict__ out, int mode) {
    int t = blockIdx.x * blockDim.x + threadIdx.x;   // 0..1023 = (b,k)
    if (t >= BATCH * NK) return;
    float* sp = s + t * DOUT;
    float sv[DOUT];
    float sn = 0.0f;
#pragma unroll
    for (int e = 0; e < DOUT; ++e) { sv[e] = sp[e]; sn = fmaf(sv[e], sv[e], sn); }
    float scale = sn / ((1.0f + sn) * sqrtf(sn + EPSV));
#pragma unroll
    for (int e = 0; e < DOUT; ++e) {
        float v = sv[e] * scale;
        if (mode == 0)      vacc[t * DOUT + e] = v;
        else if (mode == 1) vacc[t * DOUT + e] += v;
        else                out[t * DOUT + e] = v;
        sp[e] = 0.0f;   // ready for next pass / next launch
    }
}

extern "C" void kernel_launch(void* const* d_in, const int* in_sizes, int n_in,
                              void* d_out, int out_size, void* d_ws, size_t ws_size,
                              hipStream_t stream) {
    const float* X  = (const float*)d_in[0];   // [64, 2048, 8]
    const float* Wp = (const float*)d_in[1];   // [2048, 16, 8, 16]
    float* out  = (float*)d_out;               // [64, 16, 16]
    float* s    = (float*)d_ws;                // 16384 floats
    float* vacc = s + BATCH * NK * DOUT;       // 16384 floats

    dim3 pgrid(NI / CHUNK_I, BATCH / BT);      // 128 x 4 workgroups

    caps_zero<<<64, 256, 0, stream>>>(s);
    // round 0: uniform coupling
    caps_pass<false><<<pgrid, 256, 0, stream>>>(X, Wp, nullptr, s);
    caps_squash<<<4, 256, 0, stream>>>(s, vacc, out, 0);   // vacc = v0, s = 0
    // round 1: logits = u_hat . v0
    caps_pass<true><<<pgrid, 256, 0, stream>>>(X, Wp, vacc, s);
    caps_squash<<<4, 256, 0, stream>>>(s, vacc, out, 1);   // vacc = v0+v1, s = 0
    // round 2: logits = u_hat . (v0+v1)
    caps_pass<true><<<pgrid, 256, 0, stream>>>(X, Wp, vacc, s);
    caps_squash<<<4, 256, 0, stream>>>(s, vacc, out, 2);   // out = v2
}